// _LBlock_54391465837133
// MI455X (gfx1250) — compile-verified
//
#include <hip/hip_runtime.h>
#include <hip/hip_bf16.h>
#include <math.h>

#define D   128
#define H   4
#define DH  32
#define EPSV 1e-5f

typedef float v2f __attribute__((ext_vector_type(2)));
typedef float v8f __attribute__((ext_vector_type(8)));

// ---------------------------------------------------------------------------
// Monotone order-preserving float <-> uint encoding for atomicMax on floats.
__device__ __forceinline__ unsigned fenc(float f) {
  unsigned u = __float_as_uint(f);
  return (u & 0x80000000u) ? ~u : (u | 0x80000000u);
}
__device__ __forceinline__ float fdec(unsigned e) {
  unsigned u = (e & 0x80000000u) ? (e & 0x7fffffffu) : ~e;
  return __uint_as_float(u);
}
__device__ __forceinline__ float lrelu(float v) { return v > 0.f ? v : 0.2f * v; }
__device__ __forceinline__ float gelu_exact(float v) {
  return 0.5f * v * (1.f + erff(v * 0.70710678118654752f));
}

// ---------------------------------------------------------------------------
// WMMA fp32 GEMM: C[n x 128] = A[n x 128] @ W[128 x 128] (+ optional bias).
// Block = 256 threads = 8 waves; wave w computes rows [128*bx+16w, +16) x all
// 128 cols as 8 accumulators of 16x16, K swept 4 per V_WMMA_F32_16X16X4_F32.
//
// W is staged in LDS pre-paired along K:  Wp[m][c] = {W[2m][c], W[2m+1][c]}
// (m = K-pair index 0..63), so each lane's B fragment is ONE aligned
// ds_load_b64 with zero repacking. A fragment (ISA 7.12.2 f32 A 16x4 layout:
// lanes 0-15 -> K=0,1; lanes 16-31 -> K=2,3) is one unconditional
// global_load_b64 from a row-clamped pointer: out-of-range lanes read a valid
// dummy row, whose garbage only reaches output rows that are masked at store.
__global__ __launch_bounds__(256) void k_gemm128(
    const float* __restrict__ A, const float* __restrict__ W,
    const float* __restrict__ bias, float* __restrict__ C, int n)
{
  __shared__ float Wp[64 * 128 * 2];  // 64KB, [m][c][pair]
  for (int i = threadIdx.x; i < 128 * 128; i += 256) {
    int r = i >> 7, c = i & (D - 1);
    // coalesced global read; LDS write stride-2 dwords -> all 64 banks
    Wp[(((r >> 1) << 7) + c) * 2 + (r & 1)] = W[i];
  }
  __syncthreads();

  const int lane = threadIdx.x & 31;
  const int wave = threadIdx.x >> 5;
  const int half = lane >> 4;       // 0: K=0,1  1: K=2,3
  const int lidx = lane & 15;
  const int rowBase = blockIdx.x * 128 + wave * 16;
  const int aRow = rowBase + lidx;
  const float* Arow = A + (size_t)(aRow < n ? aRow : 0) * D;
  const v2f* WpV = (const v2f*)Wp;

  v8f acc[8];
#pragma unroll
  for (int j = 0; j < 8; ++j)
#pragma unroll
    for (int r = 0; r < 8; ++r) acc[j][r] = 0.f;

  for (int k = 0; k < 128; k += 4) {
    const v2f a = *(const v2f*)(Arow + k + 2 * half);        // global_load_b64
    const v2f* wrow = WpV + (((k >> 1) + half) << 7);
#pragma unroll
    for (int j = 0; j < 8; ++j) {
      const v2f b = wrow[(j << 4) + lidx];                   // ds_load_b64
#if defined(__gfx1250__) && __has_builtin(__builtin_amdgcn_wmma_f32_16x16x4_f32)
      acc[j] = __builtin_amdgcn_wmma_f32_16x16x4_f32(
          false, a, false, b, (short)0, acc[j], false, false);
#else
      acc[j][0] += a.x * b.x;  // host-pass placeholder only
#endif
    }
  }

#pragma unroll
  for (int j = 0; j < 8; ++j) {
    const int col = (j << 4) + lidx;
    const float bvv = bias ? bias[col] : 0.f;
#pragma unroll
    for (int r = 0; r < 8; ++r) {
      const int row = rowBase + r + (half << 3);
      if (row < n) C[(size_t)row * D + col] = acc[j][r] + bvv;
    }
  }
}

// ---------------------------------------------------------------------------
// GAT per-node attention logits: al[n,h] = sum_d xp[n,32h+d]*a[h,d]
__global__ __launch_bounds__(256) void k_att_logits(
    const float* __restrict__ xp, const float* __restrict__ as_,
    const float* __restrict__ ad_, float* __restrict__ als,
    float* __restrict__ ald, int n)
{
  int i = blockIdx.x * 256 + threadIdx.x;
  if (i >= n * H) return;
  int node = i >> 2, h = i & 3;
  const float* row = xp + (size_t)node * D + h * DH;
  float ss = 0.f, sd = 0.f;
#pragma unroll 8
  for (int d = 0; d < DH; ++d) {
    float x = row[d];
    ss += x * as_[h * DH + d];
    sd += x * ad_[h * DH + d];
  }
  als[i] = ss;
  ald[i] = sd;
}

// GAT edge pass 1: logits + segment max (self-loops appended implicitly)
__global__ __launch_bounds__(256) void k_gat_edge_max(
    const int* __restrict__ src, const int* __restrict__ dst, int E, int nSelf,
    const float* __restrict__ als, const float* __restrict__ ald,
    float* __restrict__ lg, unsigned* __restrict__ m)
{
  int e = blockIdx.x * 256 + threadIdx.x;
  if (e >= E + nSelf) return;
  int s = (e < E) ? src[e] : (e - E);
  int d = (e < E) ? dst[e] : (e - E);
#pragma unroll
  for (int h = 0; h < H; ++h) {
    float v = lrelu(als[s * H + h] + ald[d * H + h]);
    lg[(size_t)e * H + h] = v;
    atomicMax(&m[d * H + h], fenc(v));
  }
}

// Softmax pass 2 (shared by GAT and HGT): exp + segment sum; lg -> exp(lg-m)
__global__ __launch_bounds__(256) void k_edge_exp(
    const int* __restrict__ dst, int E, int nSelf, float* __restrict__ lg,
    const unsigned* __restrict__ m, float* __restrict__ ssum)
{
  int e = blockIdx.x * 256 + threadIdx.x;
  if (e >= E + nSelf) return;
  int d = (e < E) ? dst[e] : (e - E);
#pragma unroll
  for (int h = 0; h < H; ++h) {
    float ex = expf(lg[(size_t)e * H + h] - fdec(m[d * H + h]));
    lg[(size_t)e * H + h] = ex;
    atomicAdd(&ssum[d * H + h], ex);
  }
}

// Softmax pass 3 (shared): out[dst] += (e/s) * feat[src]; 1 wave per edge,
// float4 per lane -> coalesced 512B gather + 512B of f32 atomics per edge.
__global__ __launch_bounds__(256) void k_edge_scatter(
    const int* __restrict__ src, const int* __restrict__ dst, int E, int nSelf,
    const float* __restrict__ elg, const float* __restrict__ ssum,
    const float* __restrict__ feat, float* __restrict__ out)
{
  int gid = blockIdx.x * 256 + threadIdx.x;
  int e = gid >> 5;
  int lane = gid & 31;
  if (e >= E + nSelf) return;
  int s = (e < E) ? src[e] : (e - E);
  int d = (e < E) ? dst[e] : (e - E);
  int c = lane << 2;   // 4 contiguous cols per lane
  int h = c >> 5;
  float alpha = elg[(size_t)e * H + h] / ssum[d * H + h];
  const float4 x = *(const float4*)(feat + (size_t)s * D + c);
  float* o = out + (size_t)d * D + c;
  atomicAdd(o + 0, alpha * x.x);
  atomicAdd(o + 1, alpha * x.y);
  atomicAdd(o + 2, alpha * x.z);
  atomicAdd(o + 3, alpha * x.w);
}

// HGT edge logits: q[dst,h,:] . kt[src,h,:] * p_rel[h] / sqrt(DH) + seg max
__global__ __launch_bounds__(256) void k_hgt_logits(
    const int* __restrict__ src, const int* __restrict__ dst, int E,
    const float* __restrict__ q, const float* __restrict__ kt,
    const float* __restrict__ prel, float* __restrict__ lg,
    unsigned* __restrict__ m)
{
  int i = blockIdx.x * 256 + threadIdx.x;
  if (i >= E * H) return;
  int e = i >> 2, h = i & 3;
  int s = src[e], d = dst[e];
  const float4* qr = (const float4*)(q + (size_t)d * D + h * DH);
  const float4* kr = (const float4*)(kt + (size_t)s * D + h * DH);
  float acc = 0.f;
#pragma unroll
  for (int j = 0; j < 8; ++j) {
    float4 a = qr[j], b = kr[j];
    acc += a.x * b.x + a.y * b.y + a.z * b.z + a.w * b.w;
  }
  float v = acc * prel[h] * 0.17677669529663689f;  // 1/sqrt(32)
  lg[(size_t)e * H + h] = v;
  atomicMax(&m[d * H + h], fenc(v));
}

// Expand (H,32,32) relation tensor into block-diagonal 128x128 so the
// per-head einsum becomes one WMMA GEMM.
__global__ __launch_bounds__(256) void k_blockdiag(
    const float* __restrict__ rel, float* __restrict__ W)
{
  int idx = blockIdx.x * 256 + threadIdx.x;
  if (idx >= D * D) return;
  int r = idx >> 7, c = idx & (D - 1);
  int hr = r >> 5, hc = c >> 5;
  W[idx] = (hr == hc) ? rel[hr * DH * DH + (r & 31) * DH + (c & 31)] : 0.f;
}

__global__ __launch_bounds__(256) void k_add_bias2(
    float* __restrict__ x, const float* __restrict__ b0,
    const float* __restrict__ b1, int n)
{
  int idx = blockIdx.x * 256 + threadIdx.x;
  if (idx >= n * D) return;
  int c = idx & (D - 1);
  x[idx] += b0[c] + b1[c];
}

// BN stats: per-column sum / sum-of-squares partials (coalesced reads).
__global__ __launch_bounds__(128) void k_bn_partial(
    const float* __restrict__ x, int n, float* __restrict__ stats)
{
  int c = threadIdx.x;
  float s = 0.f, q = 0.f;
  for (int r = blockIdx.x; r < n; r += gridDim.x) {
    float v = x[(size_t)r * D + c];
    s += v;
    q += v * v;
  }
  atomicAdd(&stats[c], s);
  atomicAdd(&stats[D + c], q);
}

// y = lrelu(xres + gamma*(xin-mu)*rsqrt(var+eps) + beta)
__global__ __launch_bounds__(256) void k_bn_res_lrelu(
    const float* __restrict__ xres, const float* __restrict__ xin,
    const float* __restrict__ stats, const float* __restrict__ g,
    const float* __restrict__ b, float* __restrict__ y, int n, float invN)
{
  int idx = blockIdx.x * 256 + threadIdx.x;
  if (idx >= n * D) return;
  int c = idx & (D - 1);
  float mu = stats[c] * invN;
  float var = stats[D + c] * invN - mu * mu;
  float v = g[c] * (xin[idx] - mu) * rsqrtf(var + EPSV) + b[c];
  y[idx] = lrelu(xres[idx] + v);
}

__global__ __launch_bounds__(256) void k_gelu(float* __restrict__ x, int n)
{
  int idx = blockIdx.x * 256 + threadIdx.x;
  if (idx >= n * D) return;
  x[idx] = gelu_exact(x[idx]);
}

// x3 = sigmoid(skip)*hout + (1-sigmoid(skip))*x2
__global__ __launch_bounds__(256) void k_skipmix(
    const float* __restrict__ hout, const float* __restrict__ x2,
    const float* __restrict__ skip, float* __restrict__ y, int n)
{
  int idx = blockIdx.x * 256 + threadIdx.x;
  if (idx >= n * D) return;
  float sk = 1.f / (1.f + expf(-skip[0]));
  y[idx] = sk * hout[idx] + (1.f - sk) * x2[idx];
}

// FiLM: single block, gb = gelu(z@W1+b1)@W2+b2 (gamma=gb[:128], beta=gb[128:])
__global__ __launch_bounds__(256) void k_film(
    const float* __restrict__ z, const float* __restrict__ W1,
    const float* __restrict__ b1, const float* __restrict__ W2,
    const float* __restrict__ b2, float* __restrict__ gb)
{
  __shared__ float zs[D];
  __shared__ float hs[2 * D];
  int t = threadIdx.x;
  if (t < D) zs[t] = z[t];
  __syncthreads();
  float acc = b1[t];
  for (int i = 0; i < D; ++i) acc += zs[i] * W1[i * 256 + t];
  hs[t] = gelu_exact(acc);
  __syncthreads();
  float acc2 = b2[t];
  for (int i = 0; i < 2 * D; ++i) acc2 += hs[i] * W2[i * 256 + t];
  gb[t] = acc2;
}

// out = gamma[c]*(x3b + inj) + beta[c]
__global__ __launch_bounds__(256) void k_final(
    const float* __restrict__ a, const float* __restrict__ b,
    const float* __restrict__ gb, float* __restrict__ out, int n)
{
  int idx = blockIdx.x * 256 + threadIdx.x;
  if (idx >= n * D) return;
  int c = idx & (D - 1);
  out[idx] = gb[c] * (a[idx] + b[idx]) + gb[D + c];
}

// ---------------------------------------------------------------------------
extern "C" void kernel_launch(void* const* d_in, const int* in_sizes, int n_in,
                              void* d_out, int out_size, void* d_ws,
                              size_t ws_size, hipStream_t stream)
{
  const float* x_cell = (const float*)d_in[0];
  const float* z_h    = (const float*)d_in[1];
  const float* x_emb  = (const float*)d_in[2];
  const int*   ei0    = (const int*)d_in[3];
  const int*   ei1    = (const int*)d_in[4];
  const float* gat_W  = (const float*)d_in[5];
  const float* gat_as = (const float*)d_in[6];
  const float* gat_ad = (const float*)d_in[7];
  const float* gat_b  = (const float*)d_in[8];
  const float* bn_g   = (const float*)d_in[9];
  const float* bn_b   = (const float*)d_in[10];
  const float* Wk     = (const float*)d_in[11];
  const float* bk     = (const float*)d_in[12];
  const float* Wq     = (const float*)d_in[13];
  const float* bq     = (const float*)d_in[14];
  const float* Wv     = (const float*)d_in[15];
  const float* bvv    = (const float*)d_in[16];
  const float* a_rel  = (const float*)d_in[17];
  const float* m_rel  = (const float*)d_in[18];
  const float* p_rel  = (const float*)d_in[19];
  const float* Wo     = (const float*)d_in[20];
  const float* bo     = (const float*)d_in[21];
  const float* skip   = (const float*)d_in[22];
  const float* injW   = (const float*)d_in[23];
  const float* injb   = (const float*)d_in[24];
  const float* fW1    = (const float*)d_in[25];
  const float* fb1    = (const float*)d_in[26];
  const float* fW2    = (const float*)d_in[27];
  const float* fb2    = (const float*)d_in[28];

  const int n  = in_sizes[0] / D;
  const int E0 = in_sizes[3] / 2;
  const int E1 = in_sizes[4] / 2;
  const int Emax = E0 > E1 ? E0 : E1;

  // ---- workspace carving (256B aligned slabs) ----
  char* p = (char*)d_ws;
  auto carve = [&](size_t bytes) -> float* {
    float* r = (float*)p;
    p += (bytes + 255) & ~(size_t)255;
    return r;
  };
  const size_t NB = (size_t)n * D * sizeof(float);
  float* B0 = carve(NB);   // xp -> k -> hgt_out
  float* B1 = carve(NB);   // gatOut -> q -> x3
  float* B2 = carve(NB);   // x1 -> vt1
  float* B3 = carve(NB);   // x2
  float* B4 = carve(NB);   // v -> x3b
  float* B5 = carve(NB);   // kt -> agg -> inj
  float* B6 = carve(NB);   // vt0
  float* lgAll = carve((size_t)(E0 + E1) * H * sizeof(float));
  float* elg   = carve((size_t)(Emax + n) * H * sizeof(float));
  float* als   = carve((size_t)n * H * sizeof(float));
  float* ald   = carve((size_t)n * H * sizeof(float));
  unsigned* menc = (unsigned*)carve((size_t)n * H * sizeof(unsigned));
  float* ssum  = carve((size_t)n * H * sizeof(float));
  float* stats = carve(2 * D * sizeof(float));
  float* Wbd   = carve(D * D * sizeof(float));
  float* gbuf  = carve(2 * D * sizeof(float));

  auto cdiv = [](int a, int b) { return (a + b - 1) / b; };
  const int gemmGrid = cdiv(n, 128);
  const int ndGrid = cdiv(n * D, 256);
  const float invN = 1.f / (float)n;

  // FiLM (independent of graph path)
  k_film<<<1, 256, 0, stream>>>(z_h, fW1, fb1, fW2, fb2, gbuf);

  // ---- 2 hetero-GAT layers ----
  for (int c = 0; c < 2; ++c) {
    const float* xin = (c == 0) ? x_cell : B2;
    float* xout = (c == 0) ? B2 : B3;
    hipMemsetAsync(B1, 0, NB, stream);
    for (int t = 0; t < 2; ++t) {
      const int Et = (t == 0) ? E0 : E1;
      const int* sp = (t == 0) ? ei0 : ei1;
      const int* dp = sp + Et;
      k_gemm128<<<gemmGrid, 256, 0, stream>>>(
          xin, gat_W + (size_t)(c * 2 + t) * D * D, nullptr, B0, n);
      k_att_logits<<<cdiv(n * H, 256), 256, 0, stream>>>(
          B0, gat_as + (c * 2 + t) * H * DH, gat_ad + (c * 2 + t) * H * DH,
          als, ald, n);
      hipMemsetAsync(menc, 0, (size_t)n * H * 4, stream);  // 0 < fenc(-inf)
      hipMemsetAsync(ssum, 0, (size_t)n * H * 4, stream);
      k_gat_edge_max<<<cdiv(Et + n, 256), 256, 0, stream>>>(
          sp, dp, Et, n, als, ald, elg, menc);
      k_edge_exp<<<cdiv(Et + n, 256), 256, 0, stream>>>(
          dp, Et, n, elg, menc, ssum);
      k_edge_scatter<<<cdiv((Et + n) * 32, 256), 256, 0, stream>>>(
          sp, dp, Et, n, elg, ssum, B0, B1);
    }
    k_add_bias2<<<ndGrid, 256, 0, stream>>>(
        B1, gat_b + (c * 2 + 0) * D, gat_b + (c * 2 + 1) * D, n);
    hipMemsetAsync(stats, 0, 2 * D * 4, stream);
    k_bn_partial<<<256, 128, 0, stream>>>(B1, n, stats);
    k_bn_res_lrelu<<<ndGrid, 256, 0, stream>>>(
        xin, B1, stats, bn_g + c * D, bn_b + c * D, xout, n, invN);
  }

  // ---- HGT layer (x2 in B3) ----
  k_gemm128<<<gemmGrid, 256, 0, stream>>>(B3, Wk, bk, B0, n);
  k_gemm128<<<gemmGrid, 256, 0, stream>>>(B3, Wq, bq, B1, n);
  k_gemm128<<<gemmGrid, 256, 0, stream>>>(B3, Wv, bvv, B4, n);
  hipMemsetAsync(menc, 0, (size_t)n * H * 4, stream);
  hipMemsetAsync(ssum, 0, (size_t)n * H * 4, stream);

  float* vts[2] = {B6, B2};
  for (int t = 0; t < 2; ++t) {
    const int Et = (t == 0) ? E0 : E1;
    const int* sp = (t == 0) ? ei0 : ei1;
    const int* dp = sp + Et;
    float* lgT = lgAll + (t == 0 ? 0 : (size_t)E0 * H);
    k_blockdiag<<<cdiv(D * D, 256), 256, 0, stream>>>(a_rel + t * H * DH * DH, Wbd);
    k_gemm128<<<gemmGrid, 256, 0, stream>>>(B0, Wbd, nullptr, B5, n);  // kt
    k_blockdiag<<<cdiv(D * D, 256), 256, 0, stream>>>(m_rel + t * H * DH * DH, Wbd);
    k_gemm128<<<gemmGrid, 256, 0, stream>>>(B4, Wbd, nullptr, vts[t], n);  // vt
    k_hgt_logits<<<cdiv(Et * H, 256), 256, 0, stream>>>(
        sp, dp, Et, B1, B5, p_rel + t * H, lgT, menc);
  }
  for (int t = 0; t < 2; ++t) {
    const int Et = (t == 0) ? E0 : E1;
    const int* dp = ((t == 0) ? ei0 : ei1) + Et;
    float* lgT = lgAll + (t == 0 ? 0 : (size_t)E0 * H);
    k_edge_exp<<<cdiv(Et, 256), 256, 0, stream>>>(dp, Et, 0, lgT, menc, ssum);
  }
  hipMemsetAsync(B5, 0, NB, stream);  // agg (kt dead)
  for (int t = 0; t < 2; ++t) {
    const int Et = (t == 0) ? E0 : E1;
    const int* sp = (t == 0) ? ei0 : ei1;
    const int* dp = sp + Et;
    float* lgT = lgAll + (t == 0 ? 0 : (size_t)E0 * H);
    k_edge_scatter<<<cdiv(Et * 32, 256), 256, 0, stream>>>(
        sp, dp, Et, 0, lgT, ssum, vts[t], B5);
  }
  k_gelu<<<ndGrid, 256, 0, stream>>>(B5, n);
  k_gemm128<<<gemmGrid, 256, 0, stream>>>(B5, Wo, bo, B0, n);  // hgt_out
  k_skipmix<<<ndGrid, 256, 0, stream>>>(B0, B3, skip, B1, n);  // x3 raw
  hipMemsetAsync(stats, 0, 2 * D * 4, stream);
  k_bn_partial<<<256, 128, 0, stream>>>(B1, n, stats);
  k_bn_res_lrelu<<<ndGrid, 256, 0, stream>>>(
      B3, B1, stats, bn_g + 2 * D, bn_b + 2 * D, B4, n, invN);  // x3b
  k_gemm128<<<gemmGrid, 256, 0, stream>>>(x_emb, injW, injb, B5, n);  // inj
  k_final<<<ndGrid, 256, 0, stream>>>(B4, B5, gbuf, (float*)d_out, n);
}